// LongformerAttentionMethod_12850542150472
// MI455X (gfx1250) — compile-verified
//
#include <hip/hip_runtime.h>

typedef __attribute__((ext_vector_type(2))) float v2f;
typedef __attribute__((ext_vector_type(8))) float v8f;

constexpr int S    = 4096;
constexpr int D    = 64;
constexpr int Wn   = 256;          // attention window
constexpr int BAND = 2 * Wn + 1;   // 513
constexpr int NT   = 33;           // 16-wide key tiles covering band offsets 0..527
constexpr int SCOLS = NT * 16;     // 528
constexpr int SP   = 556;          // padded LDS score-row stride (f32), bank-conflict free
constexpr int QP   = 68;           // padded Q row stride (f32)

__global__ __launch_bounds__(128)
void longformer_wmma_kernel(const float* __restrict__ q,
                            const float* __restrict__ k,
                            const float* __restrict__ v,
                            float* __restrict__ ctx,
                            float* __restrict__ probs)
{
    __shared__ float Qs[16 * QP];      // 4.25 KB
    __shared__ float Sc[16 * SP];      // 34.8 KB

    const int bid  = blockIdx.x;
    const int bh   = bid >> 8;         // 256 query stripes per (b,h)
    const int mt   = bid & 255;
    const int s0   = mt * 16;          // first query row of this stripe
    const int tid  = threadIdx.x;
    const int lane = tid & 31;
    const int wave = tid >> 5;
    const int half = lane >> 4;        // lane half (WMMA K/M split)
    const int l16  = lane & 15;

    const float* qb = q + (size_t)bh * S * D;
    const float* kb = k + (size_t)bh * S * D;
    const float* vb = v + (size_t)bh * S * D;

    // ---- stage Q stripe into LDS, pre-scaled by 1/sqrt(D) ----
    #pragma unroll
    for (int e = 0; e < 8; ++e) {
        int i = tid + e * 128;         // 0..1023 (16 rows x 64)
        int r = i >> 6, c = i & 63;
        Qs[r * QP + c] = qb[(size_t)(s0 + r) * D + c] * 0.125f;
    }
    __syncthreads();

    const int kstart = s0 - Wn;        // key index of band offset 0 (may be < 0)

    // ---- QK: banded scores into Sc[16][528], tiles striped over waves ----
    for (int t = wave; t < NT; t += 4) {
        v8f acc = {};
        int keyrow = kstart + t * 16 + l16;
        int rowc   = keyrow < 0 ? 0 : (keyrow >= S ? S - 1 : keyrow);
        const float* kp = kb + (size_t)rowc * D;     // B fragments: per-lane key row
        const float* qp = Qs + l16 * QP;             // A fragments: per-lane query row
        #pragma unroll
        for (int kk = 0; kk < 16; ++kk) {
            int d0 = kk * 4 + half * 2;
            v2f a = *(const v2f*)(qp + d0);          // Q[m][d0..d0+1]
            v2f b = *(const v2f*)(kp + d0);          // K[key][d0..d0+1]
            acc = __builtin_amdgcn_wmma_f32_16x16x4_f32(
                      false, a, false, b, (short)0, acc, false, false);
        }
        #pragma unroll
        for (int g = 0; g < 8; ++g) {
            int m = half ? (8 + g) : g;              // C layout: VGPR g -> rows g / g+8
            Sc[m * SP + t * 16 + l16] = acc[g];
        }
    }
    __syncthreads();

    // ---- softmax over the band (8 lanes per query row) ----
    {
        const int r  = tid >> 3;
        const int l8 = tid & 7;
        float* srow = Sc + r * SP;
        const int sr = s0 + r;

        float mx = -3.0e38f;
        for (int j = l8; j < BAND; j += 8) {
            int key = sr - Wn + j;
            float x = (key >= 0 && key < S) ? srow[r + j] : -__builtin_inff();
            mx = fmaxf(mx, x);
        }
        mx = fmaxf(mx, __shfl_xor(mx, 1, 32));
        mx = fmaxf(mx, __shfl_xor(mx, 2, 32));
        mx = fmaxf(mx, __shfl_xor(mx, 4, 32));

        float sum = 0.f;
        for (int j = l8; j < BAND; j += 8) {
            int key = sr - Wn + j;
            float p = 0.f;
            if (key >= 0 && key < S) p = __expf(srow[r + j] - mx);
            srow[r + j] = p;
            sum += p;
        }
        sum += __shfl_xor(sum, 1, 32);
        sum += __shfl_xor(sum, 2, 32);
        sum += __shfl_xor(sum, 4, 32);
        float inv = 1.f / sum;

        float* prow = probs + ((size_t)bh * S + sr) * BAND;
        for (int j = l8; j < BAND; j += 8) {
            float pv = srow[r + j] * inv;
            srow[r + j] = pv;                        // in-band scatter for PV
            prow[j]     = pv;                        // attn_probs output
        }
        // zero out-of-band fringe so PV matmul sees clean zeros
        for (int o = l8; o < r; o += 8)                   srow[o] = 0.f;
        for (int o = r + BAND + l8; o < SCOLS; o += 8)    srow[o] = 0.f;
    }
    __syncthreads();

    // ---- PV: context[16 x 64] = probs_full[16 x 528] * Vwin[528 x 64] ----
    {
        const int d0 = wave * 16;                    // each wave owns 16 d-columns
        v8f acc = {};
        const float* sm = Sc + l16 * SP;             // A: per-lane prob row
        #pragma unroll 4
        for (int kk = 0; kk < SCOLS / 4; ++kk) {     // 132 k-steps
            int o = kk * 4 + half * 2;
            v2f a = *(const v2f*)(sm + o);
            int vrow = kstart + o;
            int r0 = vrow < 0 ? 0 : (vrow >= S ? S - 1 : vrow);
            int vr1 = vrow + 1;
            int r1 = vr1 < 0 ? 0 : (vr1 >= S ? S - 1 : vr1);
            v2f b;
            b.x = vb[(size_t)r0 * D + d0 + l16];     // B VGPR0: V rows {0,2}
            b.y = vb[(size_t)r1 * D + d0 + l16];     // B VGPR1: V rows {1,3}
            acc = __builtin_amdgcn_wmma_f32_16x16x4_f32(
                      false, a, false, b, (short)0, acc, false, false);
        }
        float* cb = ctx + ((size_t)bh * S + s0) * D;
        #pragma unroll
        for (int g = 0; g < 8; ++g) {
            int m = half ? (8 + g) : g;
            cb[m * D + d0 + l16] = acc[g];
        }
    }
}

extern "C" void kernel_launch(void* const* d_in, const int* in_sizes, int n_in,
                              void* d_out, int out_size, void* d_ws, size_t ws_size,
                              hipStream_t stream) {
    const float* q = (const float*)d_in[0];
    const float* k = (const float*)d_in[1];
    const float* v = (const float*)d_in[2];
    float* ctx   = (float*)d_out;
    float* probs = ctx + (size_t)2 * 16 * 4096 * 64;   // context first, then attn_probs

    dim3 grid(32 * 256);   // (B*H) x (S/16) query stripes
    longformer_wmma_kernel<<<grid, 128, 0, stream>>>(q, k, v, ctx, probs);
}